// KalmanFilter_57939108823636
// MI455X (gfx1250) — compile-verified
//
#include <hip/hip_runtime.h>

typedef __attribute__((ext_vector_type(2))) float v2f;
typedef __attribute__((ext_vector_type(8))) float v8f;

constexpr int G_ = 64;   // groups
constexpr int T_ = 256;  // time steps
constexpr int S_ = 64;   // state dim
constexpr int M_ = 8;    // measurement dim
constexpr int LDP = 68;  // padded LDS row stride: 272B rows -> 16B aligned,
                         // bank = (4*row + k) % 64 -> conflict-free columns

// ---- WMMA helpers: 64x64 f32 matmuls via V_WMMA_F32_16X16X4_F32 -------------
// A-matrix 16x4 f32 layout: lanes 0-15 hold rows M=0..15; VGPR0 = K=0 (lanes
// 0-15) / K=2 (lanes 16-31), VGPR1 = K=1 / K=3.  C/D 16x16 f32: VGPR r -> row
// r (lanes 0-15) / row r+8 (lanes 16-31), col = lane&15.

__device__ inline void store_tile(float* __restrict__ C, int ti, int tj,
                                  int lane, const v8f& acc) {
  const int col = tj * 16 + (lane & 15);
  const int rbase = ti * 16 + ((lane >> 4) << 3);
#pragma unroll
  for (int r = 0; r < 8; ++r) C[(rbase + r) * LDP + col] = acc[r];
}

// C = A @ B  (all 64x64 in LDS, row stride LDP). 8 waves, 2 tiles each.
__device__ inline void wmma_mm(const float* __restrict__ A,
                               const float* __restrict__ B,
                               float* __restrict__ C, int wave, int lane) {
#pragma unroll
  for (int ts = 0; ts < 2; ++ts) {
    const int tile = wave * 2 + ts;
    const int ti = tile >> 2, tj = tile & 3;
    const int arow = ti * 16 + (lane & 15);
    const int bcol = tj * 16 + (lane & 15);
    const int koff = (lane >> 4) << 1;
    v8f acc = {};
#pragma unroll
    for (int k = 0; k < 64; k += 4) {
      const int ka = k + koff;
      v2f a, b;
      a.x = A[arow * LDP + ka];
      a.y = A[arow * LDP + ka + 1];
      b.x = B[ka * LDP + bcol];
      b.y = B[(ka + 1) * LDP + bcol];
      acc = __builtin_amdgcn_wmma_f32_16x16x4_f32(false, a, false, b,
                                                  (short)0, acc, false, false);
    }
    store_tile(C, ti, tj, lane, acc);
  }
}

// C = A @ B^T + Qg ; store to LDS C (stride LDP) and to global outp (stride 64)
__device__ inline void wmma_mm_bt_addq(const float* __restrict__ A,
                                       const float* __restrict__ Bt,
                                       const float* __restrict__ Qg,
                                       float* __restrict__ C,
                                       float* __restrict__ outp,
                                       int wave, int lane) {
#pragma unroll
  for (int ts = 0; ts < 2; ++ts) {
    const int tile = wave * 2 + ts;
    const int ti = tile >> 2, tj = tile & 3;
    const int arow = ti * 16 + (lane & 15);
    const int bcol = tj * 16 + (lane & 15);
    const int koff = (lane >> 4) << 1;
    v8f acc = {};
#pragma unroll
    for (int k = 0; k < 64; k += 4) {
      const int ka = k + koff;
      v2f a, b;
      a.x = A[arow * LDP + ka];
      a.y = A[arow * LDP + ka + 1];
      // B[k][n] = F[n][k]  (transposed operand read; contiguous pair)
      b.x = Bt[bcol * LDP + ka];
      b.y = Bt[bcol * LDP + ka + 1];
      acc = __builtin_amdgcn_wmma_f32_16x16x4_f32(false, a, false, b,
                                                  (short)0, acc, false, false);
    }
    const int col = tj * 16 + (lane & 15);
    const int rbase = ti * 16 + ((lane >> 4) << 3);
#pragma unroll
    for (int r = 0; r < 8; ++r) {
      const int row = rbase + r;
      const float v = acc[r] + Qg[row * 64 + col];
      C[row * LDP + col] = v;
      outp[row * 64 + col] = v;
    }
  }
}

// Async global->LDS copy of one 16B chunk (CDNA5 GLOBAL_LOAD_ASYNC_TO_LDS,
// ASYNCcnt-tracked). GV addressing: vdst = LDS byte offset, v[addr] = 64-bit
// global address. Generic pointers to __shared__ carry the LDS offset in
// their low 32 bits (aperture scheme, ISA 10.2).
__device__ inline void async_copy_b128(const float* lds_dst,
                                       const float* gsrc) {
  const unsigned loff = (unsigned)(unsigned long long)lds_dst;
  asm volatile("global_load_async_to_lds_b128 %0, %1, off"
               :: "v"(loff), "v"(gsrc)
               : "memory");
}

__device__ inline void wait_async0() {
  asm volatile("s_wait_asynccnt 0x0" ::: "memory");
}

// -----------------------------------------------------------------------------
__launch_bounds__(256, 1)
__global__ void kalman_scan_kernel(const float* __restrict__ inp,  // G,T,M
                                   const float* __restrict__ Fg,   // T,S,S
                                   const float* __restrict__ Qg,   // T,S,S
                                   const float* __restrict__ Hg,   // T,M,S
                                   const float* __restrict__ Rg,   // T,M,M
                                   const float* __restrict__ m0g,  // G,S
                                   const float* __restrict__ P0g,  // G,S,S
                                   float* __restrict__ out_sm,     // G,T,S
                                   float* __restrict__ out_sc,     // G,T,S,S
                                   float* __restrict__ out_mm,     // G,T,M
                                   float* __restrict__ out_mc) {   // G,T,M,M
  __shared__ float bufA[S_ * LDP];
  __shared__ float bufB[S_ * LDP];
  __shared__ float Fsh[S_ * LDP];
  __shared__ float PHt[S_ * M_];   // [s][m]
  __shared__ float Kt[M_ * S_];    // [m][s]
  __shared__ float Hp[M_ * S_];
  __shared__ float Hc[M_ * S_];
  __shared__ float Sy[M_ * M_];
  __shared__ float SyI[M_ * M_];
  __shared__ float msh[S_];
  __shared__ float mu[S_];
  __shared__ float resid[M_];

  const int g = blockIdx.x;
  const int tid = threadIdx.x;
  const int lane = tid & 31;
  const int wave = tid >> 5;

  float* Pcur = bufA;  // holds P
  float* Pnxt = bufB;  // scratch / next P

  // ---- init: m0, P0, H[0] ---------------------------------------------------
  if (tid < S_) msh[tid] = m0g[(size_t)g * S_ + tid];
  for (int i = tid; i < S_ * S_; i += 256) {
    const int r = i >> 6, c = i & 63;
    Pcur[r * LDP + c] = P0g[(size_t)g * S_ * S_ + i];
  }
  for (int i = tid; i < M_ * S_; i += 256) Hc[i] = Hg[i];
  __syncthreads();

  // ---- t = 0 outputs --------------------------------------------------------
  if (tid < S_) out_sm[((size_t)g * T_) * S_ + tid] = msh[tid];
  for (int i = tid; i < S_ * S_; i += 256) {
    const int r = i >> 6, c = i & 63;
    out_sc[((size_t)g * T_) * S_ * S_ + i] = Pcur[r * LDP + c];
  }
  for (int i = tid; i < S_ * M_; i += 256) {   // PHt = P H0^T
    const int s = i >> 3, m = i & 7;
    float acc = 0.f;
    for (int kk = 0; kk < S_; ++kk) acc += Pcur[s * LDP + kk] * Hc[m * S_ + kk];
    PHt[i] = acc;
  }
  if (tid < M_) {                               // ym0 = H0 m0
    float acc = 0.f;
    for (int kk = 0; kk < S_; ++kk) acc += Hc[tid * S_ + kk] * msh[kk];
    out_mm[((size_t)g * T_) * M_ + tid] = acc;
  }
  __syncthreads();
  if (tid < M_ * M_) {                          // yS0 = H0 PHt + R0
    const int m = tid >> 3, n = tid & 7;
    float acc = Rg[tid];
    for (int s = 0; s < S_; ++s) acc += Hc[m * S_ + s] * PHt[s * M_ + n];
    out_mc[((size_t)g * T_) * M_ * M_ + tid] = acc;
  }
  __syncthreads();

  // ---- scan over time -------------------------------------------------------
#pragma unroll 1
  for (int t = 0; t < T_ - 1; ++t) {
    // Fire async F[t] -> Fsh copy now; consumed only after the whole
    // measurement update (hidden behind ~all the VALU work below).
    // 1024 chunks of 16B; 4 per thread. LDS rows are 16B aligned (LDP=68).
    {
      const float* fsrc = Fg + (size_t)t * S_ * S_;
#pragma unroll
      for (int j = 0; j < 4; ++j) {
        const int c = tid + 256 * j;           // chunk id 0..1023
        const int row = c >> 4;                // 4 floats per chunk, 16/row
        const int col4 = (c & 15) << 2;
        async_copy_b128(&Fsh[row * LDP + col4], fsrc + (c << 2));
      }
    }
    // Warm L2/L0 for the Q[t] reads done in the second WMMA epilogue:
    // one 64B line per thread covers the full 16KB tile.
    __builtin_prefetch(Qg + (size_t)t * S_ * S_ + tid * 16, 0, 1);

    // stage Hp=H[t], Hc=H[t+1] (small, sync; needed immediately)
    for (int i = tid; i < M_ * S_; i += 256) {
      Hp[i] = Hg[(size_t)t * M_ * S_ + i];
      Hc[i] = Hg[(size_t)(t + 1) * M_ * S_ + i];
    }
    __syncthreads();

    // resid = y - Hp m
    if (tid < M_) {
      float acc = inp[((size_t)g * T_ + t) * M_ + tid];
      for (int kk = 0; kk < S_; ++kk) acc -= Hp[tid * S_ + kk] * msh[kk];
      resid[tid] = acc;
    }
    // PHt = P Hp^T
    for (int i = tid; i < S_ * M_; i += 256) {
      const int s = i >> 3, m = i & 7;
      float acc = 0.f;
      for (int kk = 0; kk < S_; ++kk) acc += Pcur[s * LDP + kk] * Hp[m * S_ + kk];
      PHt[i] = acc;
    }
    __syncthreads();
    // Sy = Hp PHt + R
    if (tid < M_ * M_) {
      const int m = tid >> 3, n = tid & 7;
      float acc = Rg[(size_t)t * M_ * M_ + tid];
      for (int s = 0; s < S_; ++s) acc += Hp[m * S_ + s] * PHt[s * M_ + n];
      Sy[tid] = acc;
    }
    __syncthreads();
    // 8x8 Gauss-Jordan inverse (Sy is SPD) — serial, negligible cost
    if (tid == 0) {
      float a[M_][2 * M_];
      for (int r = 0; r < M_; ++r)
        for (int c = 0; c < M_; ++c) {
          a[r][c] = Sy[r * M_ + c];
          a[r][M_ + c] = (r == c) ? 1.f : 0.f;
        }
      for (int p = 0; p < M_; ++p) {
        const float pv = 1.f / a[p][p];
        for (int c = 0; c < 2 * M_; ++c) a[p][c] *= pv;
        for (int r = 0; r < M_; ++r) {
          if (r == p) continue;
          const float f = a[r][p];
          for (int c = 0; c < 2 * M_; ++c) a[r][c] -= f * a[p][c];
        }
      }
      for (int r = 0; r < M_; ++r)
        for (int c = 0; c < M_; ++c) SyI[r * M_ + c] = a[r][M_ + c];
    }
    __syncthreads();
    // Kt[m][s] = sum_n SyI[m][n] * PHt[s][n]   (Kt = Sy^-1 (PHt)^T)
    for (int i = tid; i < M_ * S_; i += 256) {
      const int m = i >> 6, s = i & 63;
      float acc = 0.f;
#pragma unroll
      for (int n = 0; n < M_; ++n) acc += SyI[m * M_ + n] * PHt[s * M_ + n];
      Kt[m * S_ + s] = acc;
    }
    __syncthreads();
    // m_u = m + Kt^T resid ; P_u = P - PHt Kt  -> Pnxt
    if (tid < S_) {
      float acc = msh[tid];
#pragma unroll
      for (int m = 0; m < M_; ++m) acc += Kt[m * S_ + tid] * resid[m];
      mu[tid] = acc;
    }
    for (int i = tid; i < S_ * S_; i += 256) {
      const int s = i >> 6, u = i & 63;
      float acc = Pcur[s * LDP + u];
#pragma unroll
      for (int m = 0; m < M_; ++m) acc -= PHt[s * M_ + m] * Kt[m * S_ + u];
      Pnxt[s * LDP + u] = acc;
    }
    wait_async0();        // F[t] tile now resident in Fsh (own wave's chunks)
    __syncthreads();      // ... and everyone else's

    // T1 = F @ P_u   (WMMA f32) -> overwrite dead Pcur
    wmma_mm(Fsh, Pnxt, Pcur, wave, lane);
    // m_p = F @ m_u  (overlapped with WMMA on waves 0-1)
    if (tid < S_) {
      float acc = 0.f;
      for (int kk = 0; kk < S_; ++kk) acc += Fsh[tid * LDP + kk] * mu[kk];
      msh[tid] = acc;
    }
    __syncthreads();

    // P_p = T1 @ F^T + Q[t]  (WMMA f32) -> Pnxt, also streamed to out_sc[t+1]
    wmma_mm_bt_addq(Pcur, Fsh, Qg + (size_t)t * S_ * S_, Pnxt,
                    out_sc + ((size_t)g * T_ + t + 1) * S_ * S_, wave, lane);
    __syncthreads();

    { float* tmp = Pcur; Pcur = Pnxt; Pnxt = tmp; }  // Pcur = P_p (uniform)

    // outputs at t+1: state mean, measurement mean/cov with Hc=H[t+1]
    if (tid < S_) out_sm[((size_t)g * T_ + t + 1) * S_ + tid] = msh[tid];
    for (int i = tid; i < S_ * M_; i += 256) {  // PHt = P_p Hc^T
      const int s = i >> 3, m = i & 7;
      float acc = 0.f;
      for (int kk = 0; kk < S_; ++kk) acc += Pcur[s * LDP + kk] * Hc[m * S_ + kk];
      PHt[i] = acc;
    }
    if (tid < M_) {
      float acc = 0.f;
      for (int kk = 0; kk < S_; ++kk) acc += Hc[tid * S_ + kk] * msh[kk];
      out_mm[((size_t)g * T_ + t + 1) * M_ + tid] = acc;
    }
    __syncthreads();
    if (tid < M_ * M_) {
      const int m = tid >> 3, n = tid & 7;
      float acc = Rg[(size_t)(t + 1) * M_ * M_ + tid];
      for (int s = 0; s < S_; ++s) acc += Hc[m * S_ + s] * PHt[s * M_ + n];
      out_mc[((size_t)g * T_ + t + 1) * M_ * M_ + tid] = acc;
    }
    __syncthreads();
  }
}

extern "C" void kernel_launch(void* const* d_in, const int* in_sizes, int n_in,
                              void* d_out, int out_size, void* d_ws,
                              size_t ws_size, hipStream_t stream) {
  (void)in_sizes; (void)n_in; (void)out_size; (void)d_ws; (void)ws_size;
  const float* inp = (const float*)d_in[0];
  const float* F   = (const float*)d_in[1];
  const float* Q   = (const float*)d_in[2];
  const float* H   = (const float*)d_in[3];
  const float* R   = (const float*)d_in[4];
  const float* m0  = (const float*)d_in[5];
  const float* P0  = (const float*)d_in[6];

  float* out    = (float*)d_out;
  float* out_sm = out;                                    // G*T*S
  float* out_sc = out_sm + (size_t)G_ * T_ * S_;          // G*T*S*S
  float* out_mm = out_sc + (size_t)G_ * T_ * S_ * S_;     // G*T*M
  float* out_mc = out_mm + (size_t)G_ * T_ * M_;          // G*T*M*M

  kalman_scan_kernel<<<G_, 256, 0, stream>>>(inp, F, Q, H, R, m0, P0,
                                             out_sm, out_sc, out_mm, out_mc);
}